// ElectricOverflow_65292092834417
// MI455X (gfx1250) — compile-verified
//
#include <hip/hip_runtime.h>
#include <hip/hip_bf16.h>

// ---------------- problem constants (match reference) ----------------
#define NM_       1000000
#define NT_       10000
#define NF_       200000
#define N_TOTAL_  (NM_ + NT_ + NF_)     // 1,210,000
#define NBX_      512
#define NBY_      512
#define NB_       (NBX_ * NBY_)         // 262,144 bins
#define BSX_      2.0f
#define BSY_      2.0f
#define INV_BS_   0.5f
#define TD_       0.9f
#define BIN_AREA_ 4.0f
#define TDA_      3.6f                  // TARGET_DENSITY * bin_area
#define SQ2B_     2.8284271247461903f   // BSX * sqrt(2)
#define K_MOV_    5
#define K_FIX_    10

typedef __attribute__((ext_vector_type(2))) float v2f;
typedef __attribute__((ext_vector_type(8))) float v8f;

// ---------------- kernel 0: zero map + accumulators ----------------
__global__ void eo_zero_kernel(float* __restrict__ dmap, float* __restrict__ accum) {
    int i = blockIdx.x * blockDim.x + threadIdx.x;
    if (i < NB_) dmap[i] = 0.0f;
    if (i == 0) {
        accum[0] = 0.0f;                 // density cost sum
        ((int*)accum)[1] = 0;            // max bits (all values >= 0)
    }
}

// ---------------- splat helper: exact rect/bin overlap scatter-add ----------------
template <int K>
__device__ __forceinline__ void eo_splat(float x, float y, float sx, float sy,
                                         float w, float* __restrict__ dmap) {
    const int bx0 = (int)floorf(x * INV_BS_);
    const int by0 = (int)floorf(y * INV_BS_);
    float ox[K], oy[K];
#pragma unroll
    for (int k = 0; k < K; ++k) {
        int   bx = bx0 + k;
        float o  = fminf(x + sx, (float)(bx + 1) * BSX_) - fmaxf(x, (float)bx * BSX_);
        o = fmaxf(o, 0.0f);
        if (bx < 0 || bx >= NBX_) o = 0.0f;
        ox[k] = o;

        int by = by0 + k;
        o = fminf(y + sy, (float)(by + 1) * BSY_) - fmaxf(y, (float)by * BSY_);
        o = fmaxf(o, 0.0f);
        if (by < 0 || by >= NBY_) o = 0.0f;
        oy[k] = o;
    }
#pragma unroll
    for (int i = 0; i < K; ++i) {
        float wox = w * ox[i];
        if (wox > 0.0f) {
            float* row = dmap + (bx0 + i) * NBY_ + by0;
#pragma unroll
            for (int j = 0; j < K; ++j) {
                float v = wox * oy[j];
                if (v > 0.0f) atomicAdd(row + j, v);   // non-returning global_atomic_add_f32
            }
        }
    }
}

// ---------------- kernel 1: splat all cells ----------------
__global__ void eo_splat_kernel(const float* __restrict__ pos,
                                const float* __restrict__ nsx,
                                const float* __restrict__ nsy,
                                float* __restrict__ dmap) {
    int i = blockIdx.x * blockDim.x + threadIdx.x;
    if (i >= N_TOTAL_) return;

    // streaming reads; hint the prefetcher ahead (emits global_prefetch_b8)
    __builtin_prefetch(pos + i + 8192, 0, 0);
    __builtin_prefetch(nsx + i + 8192, 0, 0);
    __builtin_prefetch(nsy + i + 8192, 0, 0);

    float x  = pos[i];
    float y  = pos[i + N_TOTAL_];
    float sx = nsx[i];
    float sy = nsy[i];

    const bool is_fixed = (i >= NM_) && (i < NM_ + NT_);
    if (is_fixed) {
        // init_map splat (w=1) pre-scaled by TARGET_DENSITY
        eo_splat<K_FIX_>(x, y, sx, sy, TD_, dmap);
    } else {
        float sxc = fmaxf(sx, SQ2B_);
        float syc = fmaxf(sy, SQ2B_);
        float w   = (sx * sy) / (sxc * syc);
        x += (sx - sxc) * 0.5f;
        y += (sy - syc) * 0.5f;
        eo_splat<K_MOV_>(x + 0.0f, y + 0.0f, sxc, syc, w, dmap);
    }
}

// per-bin value with border padding applied
__device__ __forceinline__ float eo_binval(const float* __restrict__ dmap, int b) {
    int ix = b >> 9;          // b = ix*NBY + iy, NBY = 512
    int iy = b & (NBY_ - 1);
    float v = dmap[b];
    bool pad = (ix == 0) | (ix == NBX_ - 1) | (iy == 0) | (iy == NBY_ - 1);
    return pad ? TDA_ : v;
}

// ---------------- kernel 2: WMMA-based sum + max reduction ----------------
// Sum trick: feed 64 clipped values/iter as the 16x4 f32 A-matrix of
// V_WMMA_F32_16X16X4_F32 against an all-ones B; D accumulates row sums
// replicated across columns, so column 0 of D = total.
__global__ void eo_reduce_kernel(const float* __restrict__ dmap, float* __restrict__ accum) {
    const int lane   = threadIdx.x & 31;
    const int wave   = (blockIdx.x * blockDim.x + threadIdx.x) >> 5;
    const int nwaves = (gridDim.x * blockDim.x) >> 5;

    v8f acc  = {0.f, 0.f, 0.f, 0.f, 0.f, 0.f, 0.f, 0.f};
    v2f ones = {1.0f, 1.0f};
    float mx = 0.0f;

    // NB_ = 262144 is a multiple of 64, so every lane is always in range:
    // EXEC stays all-ones as WMMA requires.
    for (int base = wave * 64; base < NB_; base += nwaves * 64) {
        float v0 = eo_binval(dmap, base + lane);
        float v1 = eo_binval(dmap, base + 32 + lane);
        mx = fmaxf(mx, fmaxf(v0, v1));
        v2f a = {fmaxf(v0 - TDA_, 0.0f), fmaxf(v1 - TDA_, 0.0f)};
        acc = __builtin_amdgcn_wmma_f32_16x16x4_f32(
            /*neg_a=*/false, a, /*neg_b=*/false, ones,
            /*c_mod=*/(short)0, acc, /*reuse_a=*/false, /*reuse_b=*/false);
    }

    // column reduction: lane l holds D[m=vgpr(+8 if lane>=16)][n=l%16]
    float s = acc[0] + acc[1] + acc[2] + acc[3] + acc[4] + acc[5] + acc[6] + acc[7];
    s += __shfl_xor(s, 16, 32);                 // column total == grand total
    for (int off = 16; off; off >>= 1)
        mx = fmaxf(mx, __shfl_xor(mx, off, 32));

    if (lane == 0) {
        atomicAdd(accum, s);
        atomicMax((int*)accum + 1, __float_as_int(mx));  // all values >= 0
    }
}

// ---------------- kernel 3: finalize ----------------
__global__ void eo_finalize_kernel(const float* __restrict__ accum, float* __restrict__ out) {
    out[0] = accum[0];
    out[1] = __int_as_float(((const int*)accum)[1]) * (1.0f / BIN_AREA_);
}

extern "C" void kernel_launch(void* const* d_in, const int* in_sizes, int n_in,
                              void* d_out, int out_size, void* d_ws, size_t ws_size,
                              hipStream_t stream) {
    const float* pos = (const float*)d_in[0];   // [2*N_TOTAL] : x then y
    const float* nsx = (const float*)d_in[1];   // [N_TOTAL]
    const float* nsy = (const float*)d_in[2];   // [N_TOTAL]
    float* out  = (float*)d_out;                // [2] : density_cost, max_density
    float* dmap = (float*)d_ws;                 // NB_ floats (1 MB)
    float* accm = dmap + NB_;                   // [0]=cost sum, [1]=max bits

    // zero map + accumulators
    eo_zero_kernel<<<(NB_ + 255) / 256, 256, 0, stream>>>(dmap, accm);

    // scatter-add splat (atomics into L2-resident 1 MB map)
    eo_splat_kernel<<<(N_TOTAL_ + 255) / 256, 256, 0, stream>>>(pos, nsx, nsy, dmap);

    // WMMA reduction: 128 blocks x 256 threads = 1024 waves, 4 sweeps of 64 bins/wave
    eo_reduce_kernel<<<128, 256, 0, stream>>>(dmap, accm);

    // write outputs
    eo_finalize_kernel<<<1, 1, 0, stream>>>(accm, out);
}